// STAR_28836410425493
// MI455X (gfx1250) — compile-verified
//
#include <hip/hip_runtime.h>
#include <hip/hip_bf16.h>

// ---------------- problem constants ----------------
constexpr int Bn  = 512;
constexpr int Vn  = 6890;
constexpr int Jn  = 24;
constexpr int M3  = Vn * 3;   // 20670 rows of the "dirs" GEMMs

typedef __attribute__((ext_vector_type(2))) float v2f;
typedef __attribute__((ext_vector_type(8))) float v8f;

__device__ __constant__ int PAR[23] = {0,0,0,1,2,3,4,5,6,7,8,9,9,9,12,13,14,16,17,18,19,20,21};

// One fp32 WMMA step: D(16x16) += A(16x4) * B(4x16)
__device__ inline v8f wmma_f32_step(v2f a, v2f b, v8f c) {
  // (neg_a, A, neg_b, B, c_mod, C, reuse_a, reuse_b)
  return __builtin_amdgcn_wmma_f32_16x16x4_f32(false, a, false, b, (short)0, c, false, false);
}

// ---------------------------------------------------------------------------
// K1/K4: out[b][d] = sum_k dirs[d][k] * feat[b][k] + addend
//   dirs: (M3 x LDK) row-major with KA valid columns
//   feat: (Bn x LDF) row-major with KF valid columns (KF=LDF => no guards)
//   MODE 0: addend[d] (v_template flat)   1: addend[b*M3+d] (v_shaped)
// One wave per 16(d) x 16(b) tile, V_WMMA_F32_16X16X4_F32, K fully unrolled.
// Row index is clamped (not predicated): clamped rows' results land in D rows
// that are never stored, so loads stay unconditional (no exec juggling).
// ---------------------------------------------------------------------------
template<int KA, int KF, int LDK, int LDF, int MODE>
__global__ __launch_bounds__(32)
void dirs_gemm_kernel(const float* __restrict__ dirs,
                      const float* __restrict__ feat,
                      const float* __restrict__ addend,
                      float* __restrict__ outp) {
  const int d0   = blockIdx.x * 16;
  const int b0   = blockIdx.y * 16;
  const int lane = threadIdx.x;         // 0..31
  const int m    = lane & 15;           // A row / B col / D col index
  const int kh   = (lane >> 4) * 2;     // K sub-offset per lane half

  int row = d0 + m;
  row = row < M3 ? row : (M3 - 1);      // clamp, no predication
  const float* dp = dirs + (size_t)row * LDK;
  const float* fp = feat + (size_t)(b0 + m) * LDF;

  v8f acc = {};
  constexpr int NS = (KA + 3) / 4;
#pragma unroll
  for (int s = 0; s < NS; ++s) {
    const int k0 = s * 4 + kh;
    const int k1 = k0 + 1;
    // Tail guards fold to plain loads for all but the last unrolled step;
    // in the tail they become clamped loads + v_cndmask (no branches).
    const float ax = dp[k0 < KA ? k0 : 0];
    const float ay = dp[k1 < KA ? k1 : 0];
    const float bx = fp[k0 < KF ? k0 : 0];
    const float by = fp[k1 < KF ? k1 : 0];
    v2f a, b;
    a.x = (k0 < KA) ? ax : 0.0f;
    a.y = (k1 < KA) ? ay : 0.0f;
    b.x = (k0 < KF) ? bx : 0.0f;
    b.y = (k1 < KF) ? by : 0.0f;
    acc = wmma_f32_step(a, b, acc);
  }

  // Stage D tile (rows = d, cols = batch) through LDS so stores are
  // contiguous in d for a fixed batch (coalesced 64B groups).
  __shared__ float tile[16][17];
  const int mb = (lane >> 4) * 8;
#pragma unroll
  for (int r = 0; r < 8; ++r) tile[mb + r][m] = acc[r];
  __syncthreads();

#pragma unroll
  for (int it = 0; it < 8; ++it) {
    const int idx = it * 32 + lane;
    const int n  = idx >> 4;      // batch within tile
    const int mm = idx & 15;      // d within tile
    const int rr = d0 + mm;
    if (rr < M3) {
      const float add = MODE ? addend[(size_t)(b0 + n) * M3 + rr]
                             : addend[rr];
      outp[(size_t)(b0 + n) * M3 + rr] = tile[mm][n] + add;
    }
  }
}

// ---------------------------------------------------------------------------
// K2: Jp[b][j][c] = sum_i J_regressor[j][i] * v_shaped[b][i][c]
// ---------------------------------------------------------------------------
__global__ __launch_bounds__(256)
void jreg_kernel(const float* __restrict__ Jreg,
                 const float* __restrict__ vshaped,
                 float* __restrict__ Jp) {
  const int j = blockIdx.x;
  const int b = blockIdx.y;
  const int t = threadIdx.x;
  const float* vs = vshaped + (size_t)b * M3;
  const float* jr = Jreg + (size_t)j * Vn;
  float a0 = 0.f, a1 = 0.f, a2 = 0.f;
  for (int i = t; i < Vn; i += 256) {
    const float w = jr[i];
    a0 += w * vs[i * 3 + 0];
    a1 += w * vs[i * 3 + 1];
    a2 += w * vs[i * 3 + 2];
  }
  __shared__ float red[256][3];
  red[t][0] = a0; red[t][1] = a1; red[t][2] = a2;
  __syncthreads();
  for (int off = 128; off > 0; off >>= 1) {
    if (t < off) {
      red[t][0] += red[t + off][0];
      red[t][1] += red[t + off][1];
      red[t][2] += red[t + off][2];
    }
    __syncthreads();
  }
  if (t == 0) {
    float* o = Jp + ((size_t)b * Jn + j) * 3;
    o[0] = red[0][0]; o[1] = red[0][1]; o[2] = red[0][2];
  }
}

// ---------------------------------------------------------------------------
// K3: per-batch (one wave): quaternion features, Rodrigues, kinematic chain,
//     pose_feat (padded to 96), G_skin (24x16), J_transformed.
// ---------------------------------------------------------------------------
__global__ __launch_bounds__(32)
void chain_kernel(const float* __restrict__ pose,
                  const float* __restrict__ betas,
                  const float* __restrict__ trans,
                  const float* __restrict__ Jp,
                  float* __restrict__ pose_feat,
                  float* __restrict__ gskin,
                  float* __restrict__ jt_out) {
  const int b = blockIdx.x;
  const int lane = threadIdx.x;

  __shared__ float Gloc[Jn][12];
  __shared__ float G[Jn][12];
  __shared__ float Jps[Jn][3];

  if (lane < Jn) {
    const int j = lane;
    const float t0 = pose[b * 72 + j * 3 + 0];
    const float t1 = pose[b * 72 + j * 3 + 1];
    const float t2 = pose[b * 72 + j * 3 + 2];
    const float e0 = t0 + 1e-8f, e1 = t1 + 1e-8f, e2 = t2 + 1e-8f;
    const float angle = sqrtf(e0 * e0 + e1 * e1 + e2 * e2);
    const float inv = 1.0f / angle;
    const float nx = t0 * inv, ny = t1 * inv, nz = t2 * inv;
    const float c = cosf(angle * 0.5f);
    const float s = sinf(angle * 0.5f);
    const float x = s * nx, y = s * ny, z = s * nz, w = c;

    // pose_feat = [quat feats of joints 1..23 (92), betas[:,1] (1), pad to 96]
    if (j >= 1) {
      float* pf = pose_feat + (size_t)b * 96 + (j - 1) * 4;
      pf[0] = x; pf[1] = y; pf[2] = z; pf[3] = c - 1.0f;
    } else {
      float* pf = pose_feat + (size_t)b * 96;
      pf[92] = betas[b * 10 + 1];
      pf[93] = 0.0f; pf[94] = 0.0f; pf[95] = 0.0f;
    }

    const float xx = x * x, yy = y * y, zz = z * z;
    const float wx = w * x, wy = w * y, wz = w * z;
    const float xy = x * y, xz = x * z, yz = y * z;
    const float R00 = 1 - 2 * (yy + zz), R01 = 2 * (xy - wz), R02 = 2 * (xz + wy);
    const float R10 = 2 * (xy + wz), R11 = 1 - 2 * (xx + zz), R12 = 2 * (yz - wx);
    const float R20 = 2 * (xz - wy), R21 = 2 * (yz + wx), R22 = 1 - 2 * (xx + yy);

    const float jp0 = Jp[((size_t)b * Jn + j) * 3 + 0];
    const float jp1 = Jp[((size_t)b * Jn + j) * 3 + 1];
    const float jp2 = Jp[((size_t)b * Jn + j) * 3 + 2];
    Jps[j][0] = jp0; Jps[j][1] = jp1; Jps[j][2] = jp2;

    float r0 = jp0, r1 = jp1, r2 = jp2;
    if (j > 0) {
      const int p = PAR[j - 1];
      r0 -= Jp[((size_t)b * Jn + p) * 3 + 0];
      r1 -= Jp[((size_t)b * Jn + p) * 3 + 1];
      r2 -= Jp[((size_t)b * Jn + p) * 3 + 2];
    }
    Gloc[j][0] = R00; Gloc[j][1] = R01; Gloc[j][2]  = R02; Gloc[j][3]  = r0;
    Gloc[j][4] = R10; Gloc[j][5] = R11; Gloc[j][6]  = R12; Gloc[j][7]  = r1;
    Gloc[j][8] = R20; Gloc[j][9] = R21; Gloc[j][10] = R22; Gloc[j][11] = r2;
  }
  __syncthreads();

  if (lane == 0) {
#pragma unroll
    for (int c = 0; c < 12; ++c) G[0][c] = Gloc[0][c];
    for (int i = 1; i < Jn; ++i) {
      const int p = PAR[i - 1];
      for (int r = 0; r < 3; ++r) {
        const float p0 = G[p][r * 4 + 0], p1 = G[p][r * 4 + 1],
                    p2 = G[p][r * 4 + 2], p3 = G[p][r * 4 + 3];
        G[i][r * 4 + 0] = p0 * Gloc[i][0] + p1 * Gloc[i][4] + p2 * Gloc[i][8];
        G[i][r * 4 + 1] = p0 * Gloc[i][1] + p1 * Gloc[i][5] + p2 * Gloc[i][9];
        G[i][r * 4 + 2] = p0 * Gloc[i][2] + p1 * Gloc[i][6] + p2 * Gloc[i][10];
        G[i][r * 4 + 3] = p0 * Gloc[i][3] + p1 * Gloc[i][7] + p2 * Gloc[i][11] + p3;
      }
    }
  }
  __syncthreads();

  if (lane < Jn) {
    const int j = lane;
    const float jp0 = Jps[j][0], jp1 = Jps[j][1], jp2 = Jps[j][2];
    const float s0 = G[j][0] * jp0 + G[j][1] * jp1 + G[j][2]  * jp2;
    const float s1 = G[j][4] * jp0 + G[j][5] * jp1 + G[j][6]  * jp2;
    const float s2 = G[j][8] * jp0 + G[j][9] * jp1 + G[j][10] * jp2;
    float* gs = gskin + ((size_t)b * Jn + j) * 16;
    gs[0]  = G[j][0]; gs[1]  = G[j][1]; gs[2]  = G[j][2];  gs[3]  = G[j][3]  - s0;
    gs[4]  = G[j][4]; gs[5]  = G[j][5]; gs[6]  = G[j][6];  gs[7]  = G[j][7]  - s1;
    gs[8]  = G[j][8]; gs[9]  = G[j][9]; gs[10] = G[j][10]; gs[11] = G[j][11] - s2;
    gs[12] = 0.f; gs[13] = 0.f; gs[14] = 0.f; gs[15] = 1.f;

    // J_transformed is never re-read on device: non-temporal store.
    float* jt = jt_out + ((size_t)b * Jn + j) * 3;
    __builtin_nontemporal_store(G[j][3]  + trans[b * 3 + 0], jt + 0);
    __builtin_nontemporal_store(G[j][7]  + trans[b * 3 + 1], jt + 1);
    __builtin_nontemporal_store(G[j][11] + trans[b * 3 + 2], jt + 2);
  }
}

// ---------------------------------------------------------------------------
// K5: fused skinning. Per 16(vertex) x 16(batch) tile:
//   T(16x16) = weights(16x24) x G_skin[b](24x16)  via 6 x WMMA f32 16x16x4,
//   then apply the 3x4 transform to [v_posed,1] and add trans.
// ---------------------------------------------------------------------------
__global__ __launch_bounds__(32)
void skin_kernel(const float* __restrict__ weights,
                 const float* __restrict__ gskin,
                 const float* __restrict__ vposed,
                 const float* __restrict__ trans,
                 float* __restrict__ vout) {
  const int v0   = blockIdx.x * 16;
  const int b0   = blockIdx.y * 16;
  const int lane = threadIdx.x;
  const int m    = lane & 15;
  const int kh   = (lane >> 4) * 2;

  // Preload A = weights tile (reused across all 16 batches); clamped rows'
  // products land in D rows that are never stored.
  int row = v0 + m;
  row = row < Vn ? row : (Vn - 1);
  const float* wp = weights + (size_t)row * Jn;
  v2f a[6];
#pragma unroll
  for (int s = 0; s < 6; ++s) {
    const int k0 = s * 4 + kh;
    a[s].x = wp[k0];
    a[s].y = wp[k0 + 1];
  }

  __shared__ float tile[16][17];

  for (int nb = 0; nb < 16; ++nb) {
    const int b = b0 + nb;
    const float* G = gskin + (size_t)b * Jn * 16;
    v8f acc = {};
#pragma unroll
    for (int s = 0; s < 6; ++s) {
      const int k0 = s * 4 + kh;
      v2f bb;
      bb.x = G[k0 * 16 + m];
      bb.y = G[(k0 + 1) * 16 + m];
      acc = wmma_f32_step(a[s], bb, acc);
    }
    const int mb = (lane >> 4) * 8;
#pragma unroll
    for (int r = 0; r < 8; ++r) tile[mb + r][m] = acc[r];
    __syncthreads();

    if (lane < 16) {
      const int v = v0 + lane;
      if (v < Vn) {
        const float* vp = vposed + ((size_t)b * M3 + (size_t)v * 3);
        const float h0 = vp[0], h1 = vp[1], h2 = vp[2];
        const float* T = tile[lane];
        const float x = T[0] * h0 + T[1] * h1 + T[2]  * h2 + T[3];
        const float y = T[4] * h0 + T[5] * h1 + T[6]  * h2 + T[7];
        const float z = T[8] * h0 + T[9] * h1 + T[10] * h2 + T[11];
        // Final vertices are never re-read on device: non-temporal stores
        // keep the 42MB stream from evicting v_posed/gskin out of L2.
        float* o = vout + ((size_t)b * M3 + (size_t)v * 3);
        __builtin_nontemporal_store(x + trans[b * 3 + 0], o + 0);
        __builtin_nontemporal_store(y + trans[b * 3 + 1], o + 1);
        __builtin_nontemporal_store(z + trans[b * 3 + 2], o + 2);
      }
    }
    __syncthreads();
  }
}

// ---------------------------------------------------------------------------
extern "C" void kernel_launch(void* const* d_in, const int* in_sizes, int n_in,
                              void* d_out, int out_size, void* d_ws, size_t ws_size,
                              hipStream_t stream) {
  const float* pose        = (const float*)d_in[0];
  const float* betas       = (const float*)d_in[1];
  const float* trans       = (const float*)d_in[2];
  const float* v_template  = (const float*)d_in[3];
  const float* shapedirs   = (const float*)d_in[4];
  const float* posedirs    = (const float*)d_in[5];
  const float* J_regressor = (const float*)d_in[6];
  const float* weights     = (const float*)d_in[7];

  float* out         = (float*)d_out;
  float* v_out       = out;                              // (B,V,3)
  float* vposed_out  = out + (size_t)1 * Bn * M3;        // (B,V,3)
  float* vshaped_out = out + (size_t)2 * Bn * M3;        // (B,V,3)
  float* jt_out      = out + (size_t)3 * Bn * M3;        // (B,J,3)

  float* ws        = (float*)d_ws;
  float* pose_feat = ws;                                 // B*96
  float* Jp        = ws + (size_t)Bn * 96;               // B*72
  float* gskin     = Jp + (size_t)Bn * 72;               // B*384

  const dim3 gGemm((M3 + 15) / 16, Bn / 16);

  // 1) v_shaped = shapedirs @ betas^T + v_template   (K=10)
  dirs_gemm_kernel<10, 10, 10, 10, 0><<<gGemm, 32, 0, stream>>>(
      shapedirs, betas, v_template, vshaped_out);
  // 2) Jp = J_regressor @ v_shaped
  jreg_kernel<<<dim3(Jn, Bn), 256, 0, stream>>>(J_regressor, vshaped_out, Jp);
  // 3) pose features + kinematic chain + G_skin + J_transformed
  chain_kernel<<<Bn, 32, 0, stream>>>(pose, betas, trans, Jp,
                                      pose_feat, gskin, jt_out);
  // 4) v_posed = v_shaped + posedirs @ pose_feat^T   (K=93, feat padded to 96)
  dirs_gemm_kernel<93, 96, 93, 96, 1><<<gGemm, 32, 0, stream>>>(
      posedirs, pose_feat, vshaped_out, vposed_out);
  // 5) fused LBS skinning
  skin_kernel<<<dim3((Vn + 15) / 16, Bn / 16), 32, 0, stream>>>(
      weights, gskin, vposed_out, trans, v_out);
}